// SolvationModule_11699490915123
// MI455X (gfx1250) — compile-verified
//
#include <hip/hip_runtime.h>
#include <math.h>

typedef float v2f __attribute__((ext_vector_type(2)));
typedef float v8f __attribute__((ext_vector_type(8)));

#define CONV_F 332.07156f

// Fast HW approximations: v_sqrt_f32 / v_rcp_f32 (~1 ulp). Safe here:
// d2 >= 3e-6 (positive, normal range), D+1e-6 > 1e-3.
__device__ __forceinline__ float fast_sqrt(float x) {
    return __builtin_amdgcn_sqrtf(x);
}
__device__ __forceinline__ float fast_rcp(float x) {
    return __builtin_amdgcn_rcpf(x);
}

// ---------------------------------------------------------------------------
// Kernel 1: one workgroup per (16-row tile, batch). 8 waves; each wave sweeps
// a strided subset of the 128 column tiles. Per tile: one
// v_wmma_f32_16x16x4_f32 computes the 16x16 Gram block of X·X^T; the lane
// then forms D = sqrt(|xi|^2+|xj|^2-2G+3e-6), invD = 1/(D+1e-6) with HW
// trans ops, accumulating row-sums of invD and sum(invD*D) in registers.
// B-tile loads are software-pipelined one iteration ahead.
// ---------------------------------------------------------------------------
__global__ __launch_bounds__(256)
void solv_pair_kernel(const float* __restrict__ X,
                      const float* __restrict__ embs,
                      const float* __restrict__ qs,
                      const float* __restrict__ Born,
                      const float* __restrict__ die,
                      const float* __restrict__ sfilt,
                      float* __restrict__ ws_partial,
                      int N, int C)
{
    const int tileI   = blockIdx.x;
    const int b       = blockIdx.y;
    const int tid     = threadIdx.x;
    const int lane    = tid & 31;
    const int wave    = tid >> 5;
    const int halfsel = lane >> 4;     // 0: lanes 0-15, 1: lanes 16-31
    const int lane16  = lane & 15;
    const int i0      = tileI * 16;

    __shared__ float normA[16];
    __shared__ float rowW[8][16];      // per-wave row partials
    __shared__ float sumIDW[8];        // per-wave sum(invD*D) partials
    __shared__ float partRed[16];

    const float* Xb = X + (size_t)b * N * 3;

    // |x_i|^2 for the 16 A-rows of this workgroup
    if (tid < 16) {
        int i = i0 + tid;
        float x = Xb[i * 3 + 0], y = Xb[i * 3 + 1], z = Xb[i * 3 + 2];
        normA[tid] = x * x + y * y + z * z;
    }
    __syncthreads();

    // A fragment (16x4, K=3 zero-padded to 4), fixed for all column tiles.
    // Lanes 0-15: (K0,K1)=(x,y) of row M=lane; lanes 16-31: (K2,K3)=(z,0).
    v2f afrag;
    {
        int i = i0 + lane16;
        float ax = Xb[i * 3 + 0], ay = Xb[i * 3 + 1], az = Xb[i * 3 + 2];
        afrag.x = halfsel ? az : ax;
        afrag.y = halfsel ? 0.0f : ay;
    }

    // A-row norms this lane needs for its 8 accumulator rows (C/D layout:
    // VGPR v -> M = v + 8*halfsel).
    float nA[8];
#pragma unroll
    for (int v = 0; v < 8; ++v) nA[v] = normA[v + 8 * halfsel];

    float rowAcc[8];
#pragma unroll
    for (int v = 0; v < 8; ++v) rowAcc[v] = 0.0f;
    float sumID = 0.0f;

    const int nTiles = N >> 4;  // 128

    // Software pipeline: prefetch first B tile's coords.
    float bx, by, bz;
    {
        int j = wave * 16 + lane16;
        bx = Xb[j * 3 + 0]; by = Xb[j * 3 + 1]; bz = Xb[j * 3 + 2];
    }

    for (int t = wave; t < nTiles; t += 8) {
        // Current tile values
        float cbx = bx, cby = by, cbz = bz;

        // Prefetch next tile (if any) so the load overlaps WMMA+elementwise.
        int tn = t + 8;
        if (tn < nTiles) {
            int j = tn * 16 + lane16;
            bx = Xb[j * 3 + 0]; by = Xb[j * 3 + 1]; bz = Xb[j * 3 + 2];
        }

        float nj = cbx * cbx + cby * cby + cbz * cbz;  // |x_j|^2, N = lane16

        // B fragment (4x16): lanes = N columns, halves = K pairs.
        v2f bfrag;
        bfrag.x = halfsel ? cbz : cbx;
        bfrag.y = halfsel ? 0.0f : cby;

        v8f c = {};
        // Gram tile: c[v] = x_i . x_j  (f32, K=4 with zero pad)
        c = __builtin_amdgcn_wmma_f32_16x16x4_f32(
                false, afrag, false, bfrag, (short)0, c, false, false);

#pragma unroll
        for (int v = 0; v < 8; ++v) {
            float d2   = nA[v] + nj - 2.0f * c[v] + 3e-6f;
            float D    = fast_sqrt(d2);          // v_sqrt_f32
            float invD = fast_rcp(D + 1e-6f);    // v_rcp_f32
            rowAcc[v] += invD;
            sumID     += invD * D;
        }
    }

    // Reduce row partials across the 16 lanes of each half (same M, varying N)
#pragma unroll
    for (int v = 0; v < 8; ++v) {
        float r = rowAcc[v];
        r += __shfl_xor(r, 1, 32);
        r += __shfl_xor(r, 2, 32);
        r += __shfl_xor(r, 4, 32);
        r += __shfl_xor(r, 8, 32);
        rowAcc[v] = r;
    }
    // Full-wave reduce of sum(invD*D)
    sumID += __shfl_xor(sumID, 1, 32);
    sumID += __shfl_xor(sumID, 2, 32);
    sumID += __shfl_xor(sumID, 4, 32);
    sumID += __shfl_xor(sumID, 8, 32);
    sumID += __shfl_xor(sumID, 16, 32);

    if (lane16 == 0) {       // lane 0 -> rows 0-7, lane 16 -> rows 8-15
#pragma unroll
        for (int v = 0; v < 8; ++v) rowW[wave][v + 8 * halfsel] = rowAcc[v];
    }
    if (lane == 0) sumIDW[wave] = sumID;
    __syncthreads();

    // Combine: rowInv[i] = sum over waves (fixed order -> deterministic),
    // then E_self + 0.5*(s_i+s_j)*rowInv per row.
    if (tid < 16) {
        float r = 0.0f;
#pragma unroll
        for (int w = 0; w < 8; ++w) r += rowW[w][tid];

        int irow = i0 + tid;
        const float* e = embs + ((size_t)b * N + irow) * C;
        float adie = 1e-6f, R = 1.0f, si = 0.0f, sj = 0.0f;
        for (int k = 0; k < C; ++k) {
            float ev = e[k];
            adie += ev * die[k];
            R    += ev * Born[k];
            si   += ev * sfilt[k];
            sj   += ev * sfilt[C + k];
        }
        float q     = qs[(size_t)b * N + irow];
        float eself = -(1.0f - 1.0f / adie) * q / (R + 1e-6f);
        partRed[tid] = eself + 0.5f * (si + sj) * r;
    }
    __syncthreads();

    if (tid == 0) {
        float wd = sfilt[2 * C];
        float total = 0.0f;
#pragma unroll
        for (int k = 0; k < 16; ++k) total += partRed[k];
        float sID = 0.0f;
#pragma unroll
        for (int w = 0; w < 8; ++w) sID += sumIDW[w];
        total += 0.5f * wd * sID;
        ws_partial[(size_t)b * gridDim.x + tileI] = total;
    }
}

// ---------------------------------------------------------------------------
// Kernel 2: deterministic tree-reduce of the 128 tile partials per batch,
// apply CONV*0.01 scaling and NaN guard.
// ---------------------------------------------------------------------------
__global__ __launch_bounds__(128)
void solv_finalize(const float* __restrict__ ws_partial,
                   float* __restrict__ out, int nTiles)
{
    int b   = blockIdx.x;
    int tid = threadIdx.x;   // 128 threads == nTiles
    __shared__ float red[128];
    red[tid] = (tid < nTiles) ? ws_partial[(size_t)b * nTiles + tid] : 0.0f;
    __syncthreads();
    for (int s = 64; s > 0; s >>= 1) {
        if (tid < s) red[tid] += red[tid + s];
        __syncthreads();
    }
    if (tid == 0) {
        float E = CONV_F * red[0] * 0.01f;
        out[b] = isnan(E) ? 1e-6f : E;
    }
}

extern "C" void kernel_launch(void* const* d_in, const int* in_sizes, int n_in,
                              void* d_out, int out_size, void* d_ws, size_t ws_size,
                              hipStream_t stream)
{
    const float* X    = (const float*)d_in[0];  // (B,N,3)
    const float* embs = (const float*)d_in[1];  // (B,N,C)
    const float* qs   = (const float*)d_in[2];  // (B,N)
    // d_in[3] paired_mask: unused by the reference
    const float* Born = (const float*)d_in[4];  // (C,)
    const float* die  = (const float*)d_in[5];  // (C,)
    const float* sflt = (const float*)d_in[6];  // (2C+1, 1)

    float* out = (float*)d_out;
    float* ws  = (float*)d_ws;

    const int B = out_size;                 // 4
    const int N = in_sizes[2] / B;          // 2048
    const int C = in_sizes[4];              // 32
    const int nTiles = N / 16;              // 128

    dim3 grid(nTiles, B);
    solv_pair_kernel<<<grid, 256, 0, stream>>>(X, embs, qs, Born, die, sflt,
                                               ws, N, C);
    solv_finalize<<<B, 128, 0, stream>>>(ws, out, nTiles);
}